// AdaptiveSparseVoxels_47536698032166
// MI455X (gfx1250) — compile-verified
//
#include <hip/hip_runtime.h>
#include <hip/hip_bf16.h>

// Octree subdivision: N parents -> 8N children, child row = 32 floats
// [pos.xyz, size*0.5, density, colors[27]], plus int32 morton per child.
// Bandwidth-bound (~300 MB traffic -> ~13us @ 23.3 TB/s).
//  - async global->LDS b128 DMA stages each parent's inputs once per WG
//  - LDS broadcast feeds the x8 replication
//  - non-temporal b128 stores (constant instruction offsets) stream output

typedef float v4f __attribute__((ext_vector_type(4)));

#define PPB 32          // parents per block
#define TPB 256         // threads per block (8 waves of 32)

__device__ __forceinline__ unsigned part1by2(unsigned n) {
    n &= 1023u;
    n = (n ^ (n << 16)) & 0xFF0000FFu;   // 4278190335
    n = (n ^ (n << 8))  & 0x0300F00Fu;   // 50393103
    n = (n ^ (n << 4))  & 0x030C30C3u;   // 51130563
    n = (n ^ (n << 2))  & 0x09249249u;   // 153391689
    return n;
}

// Low 32 bits of a flat LDS pointer are the raw LDS byte offset
// (aperture bits live in addr[63:32] on gfx1250).
__device__ __forceinline__ unsigned lds_raw_offset(const void* p) {
    return (unsigned)(unsigned long long)p;
}

__device__ __forceinline__ int morton_code(float cx, float cy, float cz,
                                           float fgres, int gres) {
    int ix = (int)(((cx + 1.0f) * 0.5f) * fgres);
    int iy = (int)(((cy + 1.0f) * 0.5f) * fgres);
    int iz = (int)(((cz + 1.0f) * 0.5f) * fgres);
    ix = ix < 0 ? 0 : (ix > gres - 1 ? gres - 1 : ix);
    iy = iy < 0 ? 0 : (iy > gres - 1 ? gres - 1 : iy);
    iz = iz < 0 ? 0 : (iz > gres - 1 ? gres - 1 : iz);
    return (int)((part1by2((unsigned)iz) << 2) +
                 (part1by2((unsigned)iy) << 1) +
                  part1by2((unsigned)ix));
}

// ---------------- main kernel: full 32-parent blocks only ----------------
__global__ void __launch_bounds__(TPB)
asv_main(const float* __restrict__ positions,
         const float* __restrict__ sizes,
         const float* __restrict__ densities,
         const float* __restrict__ colors,
         const int*  __restrict__ level_p,
         float* __restrict__ out,
         int*   __restrict__ morton) {
    // LDS floats: [0..95] pos(32x3), [96..127] size, [128..159] dens,
    //             [160..1023] colors(32x27). 1024 floats = 4KB.
    __shared__ __align__(16) float sm[1024];

    const int t  = threadIdx.x;
    const int p0 = blockIdx.x * PPB;

    // ---- stage: 256 lanes x one async b128 global->LDS copy (4KB total) ----
    {
        const float* g;
        unsigned l;  // LDS float index (multiple of 4 -> 16B aligned)
        if (t < 24)      { g = positions + (size_t)p0 * 3 + 4 * t;       l = 4u * t; }
        else if (t < 32) { g = sizes     + p0 + 4 * (t - 24);            l = 96u  + 4u * (t - 24); }
        else if (t < 40) { g = densities + p0 + 4 * (t - 32);            l = 128u + 4u * (t - 32); }
        else             { g = colors + (size_t)p0 * 27 + 4 * (t - 40);  l = 160u + 4u * (t - 40); }
        const unsigned lb = lds_raw_offset(&sm[0]) + l * 4u;
        asm volatile("global_load_async_to_lds_b128 %0, %1, off"
                     :: "v"(lb), "v"(g) : "memory");
        asm volatile("s_wait_asynccnt 0" ::: "memory");
    }
    __syncthreads();

    const int   gres  = 64 << (*level_p);
    const float fgres = (float)gres;

    // Lane mapping (loop-invariant): wave row + component group
    const int row = t >> 3;          // 0..31: child row handled this iteration
    const int b   = (t & 7) * 4;     // component group 0,4,...,28
    const int j   = row & 7;         // child id (invariant across iterations)
    const int plb = row >> 3;        // parent sub-index 0..3

    const float ox = (j & 1) ? 0.25f : -0.25f;
    const float oy = (j & 2) ? 0.25f : -0.25f;
    const float oz = (j & 4) ? 0.25f : -0.25f;

    float* const obase = out + ((size_t)(p0 * 8 + row)) * 32 + b;
    int*   const mbase = morton + p0 * 8 + row;

    #pragma unroll
    for (int i = 0; i < 8; ++i) {
        const int pl = i * 4 + plb;  // local parent for this row
        v4f v;
        if (b == 0) {
            const float px = sm[3 * pl], py = sm[3 * pl + 1], pz = sm[3 * pl + 2];
            const float s  = sm[96 + pl];
            const float cx = px + ox * s;
            const float cy = py + oy * s;
            const float cz = pz + oz * s;
            v = (v4f){cx, cy, cz, s * 0.5f};
            __builtin_nontemporal_store(morton_code(cx, cy, cz, fgres, gres),
                                        mbase + i * 32);
        } else if (b == 4) {
            const int cb = 160 + 27 * pl;
            v = (v4f){sm[128 + pl], sm[cb], sm[cb + 1], sm[cb + 2]};
        } else {
            const int cb = 160 + 27 * pl + (b - 5);
            v = (v4f){sm[cb], sm[cb + 1], sm[cb + 2], sm[cb + 3]};
        }
        // 8 stores off one base with constant 4KB instruction offsets
        __builtin_nontemporal_store(v, (v4f*)(obase + (size_t)i * 1024));
    }
}

// ---------------- tail kernel: remaining <32 parents (rarely launched) ----
__global__ void __launch_bounds__(TPB)
asv_tail(const float* __restrict__ positions,
         const float* __restrict__ sizes,
         const float* __restrict__ densities,
         const float* __restrict__ colors,
         const int*  __restrict__ level_p,
         float* __restrict__ out,
         int*   __restrict__ morton,
         int pstart, int Np) {
    const int gres  = 64 << (*level_p);
    const float fgres = (float)gres;
    const int nch = (Np - pstart) * 8;
    const int idx = threadIdx.x + blockIdx.x * blockDim.x;
    if (idx >= nch) return;

    const int p = pstart + (idx >> 3);
    const int j = idx & 7;
    const float px = positions[3 * p], py = positions[3 * p + 1], pz = positions[3 * p + 2];
    const float s  = sizes[p];
    const float cx = px + ((j & 1) ? 0.25f : -0.25f) * s;
    const float cy = py + ((j & 2) ? 0.25f : -0.25f) * s;
    const float cz = pz + ((j & 4) ? 0.25f : -0.25f) * s;

    const long long r = (long long)pstart * 8 + idx;
    float* o = out + (size_t)r * 32;
    o[0] = cx; o[1] = cy; o[2] = cz; o[3] = s * 0.5f;
    o[4] = densities[p];
    #pragma unroll
    for (int k = 0; k < 27; ++k) o[5 + k] = colors[(size_t)p * 27 + k];
    morton[r] = morton_code(cx, cy, cz, fgres, gres);
}

extern "C" void kernel_launch(void* const* d_in, const int* in_sizes, int n_in,
                              void* d_out, int out_size, void* d_ws, size_t ws_size,
                              hipStream_t stream) {
    const float* positions = (const float*)d_in[0];
    const float* sizes     = (const float*)d_in[1];
    const float* densities = (const float*)d_in[2];
    const float* colors    = (const float*)d_in[3];
    const int*   level     = (const int*)d_in[4];
    float* out = (float*)d_out;

    const int Np = in_sizes[1];                     // number of parents
    int* morton = (int*)(out + (size_t)Np * 8 * 32); // second tuple output

    const int full_blocks = Np / PPB;
    if (full_blocks > 0) {
        asv_main<<<full_blocks, TPB, 0, stream>>>(
            positions, sizes, densities, colors, level, out, morton);
    }
    const int pstart = full_blocks * PPB;
    const int tail   = Np - pstart;
    if (tail > 0) {
        const int nch = tail * 8;
        asv_tail<<<(nch + TPB - 1) / TPB, TPB, 0, stream>>>(
            positions, sizes, densities, colors, level, out, morton, pstart, Np);
    }
}